// DetectionPostProcessV1_15719580304012
// MI455X (gfx1250) — compile-verified
//
#include <hip/hip_runtime.h>
#include <hip/hip_bf16.h>
#include <stdint.h>

// ---------------- problem constants ----------------
constexpr int kB = 8, kH = 128, kW = 128, kA = 8, kC = 8;
constexpr int kHW = kH * kW;            // 16384
constexpr int kN  = kHW * kA;           // 131072 anchors per batch
constexpr int PREK = 1024, POSTK = 300;
constexpr int CAPK = 16384;             // tie-collection capacity per batch
constexpr int BOX_THR = 9;
constexpr float kNmsThr = 0.5f;         // 128/256
constexpr float kClassSep = 4096.0f;

// ---------------- CDNA5 feature probes ----------------
#if defined(__AMDGCN__) && __has_builtin(__builtin_amdgcn_global_load_async_to_lds_b128)
#define HAVE_ASYNC_LDS 1
#else
#define HAVE_ASYNC_LDS 0
#endif

#if defined(__AMDGCN__) && __has_builtin(__builtin_amdgcn_wmma_f32_16x16x4_f32)
#define HAVE_WMMA4 1
#else
#define HAVE_WMMA4 0
#endif

typedef __attribute__((ext_vector_type(2))) float v2f;
typedef __attribute__((ext_vector_type(8))) float v8f;
typedef int v4i __attribute__((vector_size(16)));   // matches builtin param type

#if HAVE_ASYNC_LDS
typedef __attribute__((address_space(1))) v4i* g_v4i_p;   // global int4*
typedef __attribute__((address_space(3))) v4i* l_v4i_p;   // LDS int4*
static __device__ __forceinline__ void async_cp16(const float* gsrc, float* ldst) {
  __builtin_amdgcn_global_load_async_to_lds_b128(
      (g_v4i_p)(v4i*)const_cast<float*>(gsrc),
      (l_v4i_p)(v4i*)ldst, 0, 0);
}
#endif

static __device__ __forceinline__ unsigned ballot32(bool p) {
#if defined(__AMDGCN__) && __has_builtin(__builtin_amdgcn_ballot_w32)
  return __builtin_amdgcn_ballot_w32(p);
#else
  return (unsigned)__ballot(p);
#endif
}

// ============================================================
// K1: per-anchor class max/argmax + per-batch score histogram
// grid(kHW/256, kB), block 256. Coalesced over pixel index p.
// ============================================================
__global__ void detpp_score(const int* __restrict__ data,
                            unsigned short* __restrict__ packed,
                            unsigned* __restrict__ hist) {
  __shared__ unsigned lhist[256];
  int tid = threadIdx.x;
  for (int i = tid; i < 256; i += 256) lhist[i] = 0u;
  __syncthreads();
  int b = blockIdx.y;
  int p = blockIdx.x * 256 + tid;          // pixel index 0..16383
  for (int a = 0; a < kA; ++a) {
    const int* base = data + ((size_t)(b * kA + a) * (4 + kC)) * kHW + p;
    int best = -1000, lab = 0;
#pragma unroll
    for (int c = 0; c < kC; ++c) {
      int v = base[(4 + c) * kHW];
      if (v > best) { best = v; lab = c; }  // strict > => first argmax (jax)
    }
    int n = p * kA + a;                     // n = (h*W+w)*A + a
    packed[(size_t)b * kN + n] = (unsigned short)(((best + 128) << 8) | lab);
    if (best >= BOX_THR) atomicAdd(&lhist[best + 128], 1u);
  }
  __syncthreads();
  for (int i = tid; i < 256; i += 256)
    if (lhist[i]) atomicAdd(&hist[b * 256 + i], lhist[i]);
}

// ============================================================
// K2: per-batch threshold bin T and provisional M = min(1024, valid)
// ============================================================
__global__ void detpp_thresh(const unsigned* __restrict__ hist,
                             int* __restrict__ tbin, int* __restrict__ marr) {
  int b = blockIdx.x;
  if (threadIdx.x != 0) return;
  int total = 0;
  for (int v = 255; v >= BOX_THR + 128; --v) total += (int)hist[b * 256 + v];
  int M = total < PREK ? total : PREK;
  int T = 256;
  if (M > 0) {
    int cum = 0;
    for (int v = 255; v >= BOX_THR + 128; --v) {
      cum += (int)hist[b * 256 + v];
      if (cum >= M) { T = v; break; }
    }
  }
  tbin[b] = T;
  marr[b] = M;
}

// ============================================================
// K3: collect keys for all anchors with bin >= T.
// key = bin<<17 | (N-1-n)  => descending key == (score desc, index asc)
// ============================================================
__global__ void detpp_collect(const unsigned short* __restrict__ packed,
                              const int* __restrict__ tbin,
                              unsigned* __restrict__ cnt,
                              unsigned* __restrict__ keys) {
  int b = blockIdx.y;
  int n = blockIdx.x * 256 + threadIdx.x;
  int T = tbin[b];
  if (T > 255) return;
  unsigned pk = packed[(size_t)b * kN + n];
  int bin = (int)(pk >> 8);
  if (bin >= T) {
    unsigned pos = atomicAdd(&cnt[b], 1u);
    if (pos < (unsigned)CAPK)
      keys[(size_t)b * CAPK + pos] = ((unsigned)bin << 17) | (unsigned)(kN - 1 - n);
  }
}

// ============================================================
// K4: bitonic sort (descending) of up to 16384 keys in 64KB LDS,
// emit top-1024 (selN, topv). Deterministic regardless of atomic order.
// ============================================================
__global__ void __launch_bounds__(1024)
detpp_sort(const unsigned* __restrict__ cnt, const unsigned* __restrict__ keysg,
           int* __restrict__ marr, int* __restrict__ selN, float* __restrict__ topv) {
  __shared__ unsigned sk[CAPK];
  int b = blockIdx.x, tid = threadIdx.x;
  int c = (int)cnt[b]; if (c > CAPK) c = CAPK;
  for (int s = tid; s < CAPK; s += 1024)
    sk[s] = (s < c) ? keysg[(size_t)b * CAPK + s] : 0u;
  __syncthreads();
  for (int k = 2; k <= CAPK; k <<= 1) {
    for (int j = k >> 1; j > 0; j >>= 1) {
      for (int s = tid; s < CAPK; s += 1024) {
        int ixj = s ^ j;
        if (ixj > s) {
          bool desc = ((s & k) == 0);
          unsigned x = sk[s], y = sk[ixj];
          if (desc ? (x < y) : (x > y)) { sk[s] = y; sk[ixj] = x; }
        }
      }
      __syncthreads();
    }
  }
  int M0 = marr[b];
  int M = M0 < c ? M0 : c;
  __syncthreads();
  if (tid == 0) marr[b] = M;
  if (tid < PREK) {
    unsigned key = sk[tid];
    int idx = b * PREK + tid;
    if (tid < M) {
      selN[idx] = (int)(kN - 1 - (key & 0x1FFFFu));
      topv[idx] = (float)((int)(key >> 17) - 128) * (1.0f / 16.0f);
    } else {
      selN[idx] = -1;
      topv[idx] = -1.0e9f;
    }
  }
}

// ============================================================
// K5: decode boxes for the 1024 selected anchors (gathered deltas)
// ============================================================
__global__ void detpp_decode(const int* __restrict__ data, const float* __restrict__ anchors,
                             const int* __restrict__ exptab, const unsigned short* __restrict__ packed,
                             const int* __restrict__ selN, const int* __restrict__ marr,
                             float* __restrict__ bk, int* __restrict__ selLab,
                             float* __restrict__ ox1, float* __restrict__ oy1,
                             float* __restrict__ ox2, float* __restrict__ oy2,
                             float* __restrict__ oar) {
  int b = blockIdx.y;
  int r = blockIdx.x * 256 + threadIdx.x;
  int M = marr[b];
  int idx = b * PREK + r;
  if (r < M) {
    int n = selN[idx];
    int a = n & 7, p = n >> 3;
    const int* base = data + ((size_t)(b * kA + a) * (4 + kC)) * kHW + p;
    float dx = (float)base[0]        * (1.0f / 16.0f);
    float dy = (float)base[kHW]      * (1.0f / 16.0f);
    int dzi = base[2 * kHW], dwi = base[3 * kHW];
    float ew = (float)exptab[dzi + 128] * (1.0f / 16.0f);
    float eh = (float)exptab[dwi + 128] * (1.0f / 16.0f);
    float a0 = anchors[(size_t)n * 4 + 0], a1 = anchors[(size_t)n * 4 + 1];
    float a2 = anchors[(size_t)n * 4 + 2], a3 = anchors[(size_t)n * 4 + 3];
    float cx = (a0 + a2) * 0.5f, cy = (a1 + a3) * 0.5f;
    float w = a2 - a0, h = a3 - a1;
    float pcx = cx + dx * w, pcy = cy + dy * h;
    float pw = w * ew, ph = h * eh;
    float x1 = fminf(fmaxf(pcx - pw * 0.5f, 0.f), 1024.f);
    float y1 = fminf(fmaxf(pcy - ph * 0.5f, 0.f), 1024.f);
    float x2 = fminf(fmaxf(pcx + pw * 0.5f, 0.f), 1024.f);
    float y2 = fminf(fmaxf(pcy + ph * 0.5f, 0.f), 1024.f);
    int lab = (int)(packed[(size_t)b * kN + n] & 0xFF);
    bk[(size_t)idx * 4 + 0] = x1; bk[(size_t)idx * 4 + 1] = y1;
    bk[(size_t)idx * 4 + 2] = x2; bk[(size_t)idx * 4 + 3] = y2;
    selLab[idx] = lab;
    float off = (float)lab * kClassSep;
    ox1[idx] = x1 + off; oy1[idx] = y1 + off;
    ox2[idx] = x2 + off; oy2[idx] = y2 + off;
    oar[idx] = (x2 - x1) * (y2 - y1);
  } else {
    bk[(size_t)idx * 4 + 0] = 0.f; bk[(size_t)idx * 4 + 1] = 0.f;
    bk[(size_t)idx * 4 + 2] = 0.f; bk[(size_t)idx * 4 + 3] = 0.f;
    selLab[idx] = -1;
    ox1[idx] = 0.f; oy1[idx] = 0.f; ox2[idx] = 0.f; oy2[idx] = 0.f; oar[idx] = 0.f;
  }
}

// ============================================================
// K6: IoU > 0.5 bitmask, 1024x1024 per batch.
// - 5 SoA box arrays staged to LDS with global_load_async_to_lds_b128
// - area_i + area_j tiles computed with v_wmma_f32_16x16x4_f32
//   (A = [area_i, 1] 16x4, B = [1 ; area_j] 4x16, C = 0)
// - bits assembled with wave32 ballots
// grid(64 i-tiles, kB), block 256 (8 waves; wave handles 4 j-word tiles)
// ============================================================
__global__ void __launch_bounds__(256)
detpp_mask(const float* __restrict__ ox1, const float* __restrict__ oy1,
           const float* __restrict__ ox2, const float* __restrict__ oy2,
           const float* __restrict__ oar, unsigned* __restrict__ mask) {
  __shared__ __align__(16) float sx1[PREK];
  __shared__ __align__(16) float sy1[PREK];
  __shared__ __align__(16) float sx2[PREK];
  __shared__ __align__(16) float sy2[PREK];
  __shared__ __align__(16) float sar[PREK];
  int b = blockIdx.y;
  int it = blockIdx.x;            // i-tile (16 rows)
  int tid = threadIdx.x;
  int lane = tid & 31;
  int wav = tid >> 5;

  {
    const float* gs0 = ox1 + (size_t)b * PREK + tid * 4;
    const float* gs1 = oy1 + (size_t)b * PREK + tid * 4;
    const float* gs2 = ox2 + (size_t)b * PREK + tid * 4;
    const float* gs3 = oy2 + (size_t)b * PREK + tid * 4;
    const float* gs4 = oar + (size_t)b * PREK + tid * 4;
    float* ld0 = &sx1[tid * 4]; float* ld1 = &sy1[tid * 4];
    float* ld2 = &sx2[tid * 4]; float* ld3 = &sy2[tid * 4];
    float* ld4 = &sar[tid * 4];
#if HAVE_ASYNC_LDS
    async_cp16(gs0, ld0);
    async_cp16(gs1, ld1);
    async_cp16(gs2, ld2);
    async_cp16(gs3, ld3);
    async_cp16(gs4, ld4);
#if __has_builtin(__builtin_amdgcn_s_wait_asynccnt)
    __builtin_amdgcn_s_wait_asynccnt(0);
#else
    asm volatile("s_wait_asynccnt 0" ::: "memory");
#endif
#else
    ld0[0] = gs0[0]; ld0[1] = gs0[1]; ld0[2] = gs0[2]; ld0[3] = gs0[3];
    ld1[0] = gs1[0]; ld1[1] = gs1[1]; ld1[2] = gs1[2]; ld1[3] = gs1[3];
    ld2[0] = gs2[0]; ld2[1] = gs2[1]; ld2[2] = gs2[2]; ld2[3] = gs2[3];
    ld3[0] = gs3[0]; ld3[1] = gs3[1]; ld3[2] = gs3[2]; ld3[3] = gs3[3];
    ld4[0] = gs4[0]; ld4[1] = gs4[1]; ld4[2] = gs4[2]; ld4[3] = gs4[3];
#endif
  }
  __syncthreads();

  int i0 = it * 16;
  bool lo16 = lane < 16;
  int ibase = i0 + (lo16 ? 0 : 8);
  float x1i[8], y1i[8], x2i[8], y2i[8];
#pragma unroll
  for (int k = 0; k < 8; ++k) {
    x1i[k] = sx1[ibase + k]; y1i[k] = sy1[ibase + k];
    x2i[k] = sx2[ibase + k]; y2i[k] = sy2[ibase + k];
  }
#if HAVE_WMMA4
  v2f afrag;
  afrag.x = lo16 ? sar[i0 + lane] : 0.0f;   // K=0 col: area_i (K=2 half zero)
  afrag.y = lo16 ? 1.0f : 0.0f;             // K=1 col: ones   (K=3 half zero)
#else
  float ari[8];
#pragma unroll
  for (int k = 0; k < 8; ++k) ari[k] = sar[ibase + k];
#endif

  for (int itn = 0; itn < 4; ++itn) {
    int jt = wav + itn * 8;                 // 32-bit mask word index 0..31
    int j0 = jt * 32;
    unsigned myb0 = 0u, myb1 = 0u;
#pragma unroll
    for (int hh = 0; hh < 2; ++hh) {
      int jj0 = j0 + hh * 16;
      float dsv[8];
#if HAVE_WMMA4
      v2f bfrag;
      bfrag.x = lo16 ? 1.0f : 0.0f;              // K=0 row: ones
      bfrag.y = lo16 ? sar[jj0 + lane] : 0.0f;   // K=1 row: area_j
      v8f cz = {0.f, 0.f, 0.f, 0.f, 0.f, 0.f, 0.f, 0.f};
      v8f dsum = __builtin_amdgcn_wmma_f32_16x16x4_f32(
          false, afrag, false, bfrag, (short)0, cz, false, false);
#pragma unroll
      for (int k = 0; k < 8; ++k) dsv[k] = dsum[k];
#endif
      int j = jj0 + (lane & 15);
      float x1j = sx1[j], y1j = sy1[j], x2j = sx2[j], y2j = sy2[j];
#if !HAVE_WMMA4
      float arj = sar[j];
#pragma unroll
      for (int k = 0; k < 8; ++k) dsv[k] = ari[k] + arj;
#endif
#pragma unroll
      for (int k = 0; k < 8; ++k) {
        float iw = fminf(x2i[k], x2j) - fmaxf(x1i[k], x1j);
        float ih = fminf(y2i[k], y2j) - fmaxf(y1i[k], y1j);
        float inter = fmaxf(iw, 0.f) * fmaxf(ih, 0.f);
        float uni = dsv[k] - inter + 1e-9f;
        unsigned bal = ballot32(inter > kNmsThr * uni);
        if ((lane & 7) == k) { if (hh == 0) myb0 = bal; else myb1 = bal; }
      }
    }
    // lanes 0..7: rows i0+lane   (low 16 of each ballot)
    // lanes 8..15: rows i0+lane  (high 16 of each ballot)
    unsigned word;
    if (lane < 8) word = (myb0 & 0xFFFFu) | (myb1 << 16);
    else          word = (myb0 >> 16) | (myb1 & 0xFFFF0000u);
    if (lane < 16)
      mask[((size_t)b * PREK + (size_t)(i0 + lane)) * 32 + jt] = word;
  }
}

// ============================================================
// K7: greedy NMS scan (one wave32 per batch) + first-300 compaction
// ============================================================
__global__ void __launch_bounds__(32)
detpp_nms_out(const unsigned* __restrict__ mask, const float* __restrict__ topv,
              const float* __restrict__ bk, const int* __restrict__ selLab,
              const int* __restrict__ marr, float* __restrict__ out) {
  int b = blockIdx.x;
  int lane = threadIdx.x;            // word index 0..31
  int M = marr[b];
  int lob = lane * 32;
  unsigned remv;
  if (M <= lob)            remv = 0xFFFFFFFFu;             // all invalid
  else if (M >= lob + 32)  remv = 0u;
  else                     remv = 0xFFFFFFFFu << (M - lob);
  unsigned keep = 0u;
  int kept = 0;
  const unsigned* mrow = mask + (size_t)b * PREK * 32;
  unsigned rowv = mrow[lane];        // row 0 preloaded
  for (int i = 0; i < PREK && kept < POSTK; ++i) {
    unsigned nextv = (i + 1 < PREK) ? mrow[(size_t)(i + 1) * 32 + lane] : 0u;
    if (i + 8 < PREK) __builtin_prefetch(&mrow[(size_t)(i + 8) * 32 + lane], 0, 0);
    unsigned rw = __shfl(remv, i >> 5, 32);
    if (((rw >> (i & 31)) & 1u) == 0u) {
      if (lane == (i >> 5)) keep |= (1u << (i & 31));
      ++kept;
      remv |= rowv;
    }
    rowv = nextv;
  }
  // exclusive prefix of kept bits across lanes
  int cntk = __popc(keep);
  int pre = cntk;
  for (int d = 1; d < 32; d <<= 1) {
    int t = __shfl_up(pre, d, 32);
    if (lane >= d) pre += t;
  }
  int total = __shfl(pre, 31, 32);
  pre -= cntk;
  unsigned kk = keep;
  int p = pre;
  while (kk) {
    int bit = __ffs(kk) - 1;
    kk &= kk - 1u;
    if (p < POSTK) {
      int r = lane * 32 + bit;
      size_t src = (size_t)b * PREK + r;
      int dst = b * POSTK + p;
      out[(size_t)dst * 4 + 0] = bk[src * 4 + 0];
      out[(size_t)dst * 4 + 1] = bk[src * 4 + 1];
      out[(size_t)dst * 4 + 2] = bk[src * 4 + 2];
      out[(size_t)dst * 4 + 3] = bk[src * 4 + 3];
      out[(size_t)kB * POSTK * 4 + dst] = topv[src];
      out[(size_t)kB * POSTK * 5 + dst] = (float)selLab[src];
    }
    ++p;
  }
  int fill0 = total < POSTK ? total : POSTK;
  for (int q = fill0 + lane; q < POSTK; q += 32) {
    int dst = b * POSTK + q;
    out[(size_t)dst * 4 + 0] = 0.f; out[(size_t)dst * 4 + 1] = 0.f;
    out[(size_t)dst * 4 + 2] = 0.f; out[(size_t)dst * 4 + 3] = 0.f;
    out[(size_t)kB * POSTK * 4 + dst] = 0.f;
    out[(size_t)kB * POSTK * 5 + dst] = -1.0f;
  }
}

// ============================================================
// host-side launch
// ============================================================
extern "C" void kernel_launch(void* const* d_in, const int* in_sizes, int n_in,
                              void* d_out, int out_size, void* d_ws, size_t ws_size,
                              hipStream_t stream) {
  (void)in_sizes; (void)n_in; (void)out_size; (void)ws_size;
  const int*   data    = (const int*)d_in[0];
  const float* anchors = (const float*)d_in[1];
  const int*   exptab  = (const int*)d_in[2];
  float* out = (float*)d_out;
  char* ws = (char*)d_ws;

  size_t off = 0;
  auto alloc = [&](size_t bytes) -> void* {
    void* p = ws + off;
    off = (off + bytes + 255) & ~(size_t)255;
    return p;
  };
  unsigned* hist = (unsigned*)alloc((size_t)kB * 256 * 4);
  unsigned* cnt  = (unsigned*)alloc((size_t)kB * 4);
  int* tbin      = (int*)alloc((size_t)kB * 4);
  int* marr      = (int*)alloc((size_t)kB * 4);
  size_t headerBytes = off;
  unsigned short* packed = (unsigned short*)alloc((size_t)kB * kN * 2);
  unsigned* keys = (unsigned*)alloc((size_t)kB * CAPK * 4);
  int* selN      = (int*)alloc((size_t)kB * PREK * 4);
  float* topv    = (float*)alloc((size_t)kB * PREK * 4);
  int* selLab    = (int*)alloc((size_t)kB * PREK * 4);
  float* bk      = (float*)alloc((size_t)kB * PREK * 4 * 4);
  float* ox1     = (float*)alloc((size_t)kB * PREK * 4);
  float* oy1     = (float*)alloc((size_t)kB * PREK * 4);
  float* ox2     = (float*)alloc((size_t)kB * PREK * 4);
  float* oy2     = (float*)alloc((size_t)kB * PREK * 4);
  float* oar     = (float*)alloc((size_t)kB * PREK * 4);
  unsigned* mask = (unsigned*)alloc((size_t)kB * PREK * 32 * 4);

  (void)hipMemsetAsync(d_ws, 0, headerBytes, stream);
  detpp_score<<<dim3(kHW / 256, kB), 256, 0, stream>>>(data, packed, hist);
  detpp_thresh<<<kB, 32, 0, stream>>>(hist, tbin, marr);
  detpp_collect<<<dim3(kN / 256, kB), 256, 0, stream>>>(packed, tbin, cnt, keys);
  detpp_sort<<<kB, 1024, 0, stream>>>(cnt, keys, marr, selN, topv);
  detpp_decode<<<dim3(PREK / 256, kB), 256, 0, stream>>>(
      data, anchors, exptab, packed, selN, marr, bk, selLab, ox1, oy1, ox2, oy2, oar);
  detpp_mask<<<dim3(PREK / 16, kB), 256, 0, stream>>>(ox1, oy1, ox2, oy2, oar, mask);
  detpp_nms_out<<<kB, 32, 0, stream>>>(mask, topv, bk, selLab, marr, out);
}